// HashEmbedding2_30623116820711
// MI455X (gfx1250) — compile-verified
//
#include <hip/hip_runtime.h>
#include <stdint.h>

// Problem constants (from the reference)
#define NH     8           // N_HASH
#define DD     56          // D = DIM - N_HASH
#define BMASK  0xFFFFFu    // B_ROWS = 2^20 -> mod is a mask
#define KROWS  149796u     // K_ROWS (not a power of two)
#define WPB    8           // waves per block (256 threads)
#define NITER  8           // batch elements per wave (pipelined)

// One wave per batch element per iteration.
// Lanes 0..31 = output dims {lane, 32+lane}; lanes 24..31 also carry w[0..7]
// for the concat tail. Table rows (224 B each, 16B-aligned) are gathered into
// LDS with async global->LDS loads, double buffered across iterations.
__global__ __launch_bounds__(256)
void hashemb_gather_kernel(const int* __restrict__ xs,
                           const float* __restrict__ table,
                           const float* __restrict__ weights,
                           const int* __restrict__ a0g,
                           const int* __restrict__ b0g,
                           const int* __restrict__ a1g,
                           const int* __restrict__ b1g,
                           float* __restrict__ out,
                           int batch)
{
    // 8 waves * 2 buffers * 448 floats (1792 B) = 28 KB per block
    __shared__ float smem[WPB * 2 * NH * DD];

    const int lane = threadIdx.x & 31;
    const int wid  = threadIdx.x >> 5;
    const long gw  = (long)blockIdx.x * WPB + wid;
    const long base = gw * NITER;
    if (base >= batch) return;
    const int nit = (int)((batch - base) < NITER ? (batch - base) : NITER);

    // Hash coefficients (uniform) into registers
    uint32_t A0[NH], B0[NH];
#pragma unroll
    for (int h = 0; h < NH; ++h) {
        A0[h] = (uint32_t)a0g[h];
        B0[h] = (uint32_t)b0g[h];
    }
    const uint32_t A1 = (uint32_t)a1g[0];
    const uint32_t B1 = (uint32_t)b1g[0];

    // Low 32 bits of the flat address of a __shared__ object == LDS byte offset
    const unsigned lds_wave =
        (unsigned)(uintptr_t)(&smem[0]) + (unsigned)(wid * 2 * NH * DD * 4);
    const uint64_t tbase = (uint64_t)(uintptr_t)table;

    // ---- stage: async-gather 8 table rows (8*224 B) for element b into buffer ----
    auto stage = [&](long b, int buf) {
        const uint32_t xu = (uint32_t)xs[b];
        const unsigned ldsbuf = lds_wave + (unsigned)(buf * NH * DD * 4);
        // 112 chunks of 16 B; 4 async instructions (last one 16/32 lanes active)
#pragma unroll
        for (int p = 0; p < 4; ++p) {
            const int c = p * 32 + lane;
            if (c < (NH * DD * 4) / 16) {
                const unsigned h   = (unsigned)c / 14u;      // row 0..7
                const unsigned sub = (unsigned)c - h * 14u;  // 16B chunk in row
                const uint32_t idx = (xu * A0[h] + B0[h]) & BMASK;
                const unsigned goff = idx * (DD * 4) + sub * 16u;
                const unsigned loff = ldsbuf + (unsigned)c * 16u;
                asm volatile("global_load_async_to_lds_b128 %0, %1, %2"
                             :: "v"(loff), "v"(goff), "s"(tbase)
                             : "memory");
            }
        }
    };

    // ---- compute: mix staged rows with w, write 64 f32 (2 NT coalesced stores) ----
    auto compute = [&](long b, int buf) {
        const uint32_t xu = (uint32_t)xs[b];
        const uint32_t i1 = (xu * A1 + B1) % KROWS;
        const float* wrow = weights + (size_t)i1 * NH;
        float w[NH];
#pragma unroll
        for (int h = 0; h < NH; ++h) w[h] = wrow[h];

        const float* bf = smem + (wid * 2 + buf) * (NH * DD);
        float acc0 = 0.0f, acc1 = 0.0f;
#pragma unroll
        for (int h = 0; h < NH; ++h) {
            acc0 = fmaf(w[h], bf[h * DD + lane], acc0);
            if (lane < DD - 32)
                acc1 = fmaf(w[h], bf[h * DD + 32 + lane], acc1);
        }
        // second half: dims 32..55 then w[0..7] (concat tail)
        const float second = (lane < DD - 32) ? acc1 : w[lane - (DD - 32)];
        float* orow = out + (size_t)b * 64;
        __builtin_nontemporal_store(acc0,   orow + lane);        // 128 B
        __builtin_nontemporal_store(second, orow + 32 + lane);   // 128 B
    };

    // ---- double-buffered pipeline ----
    stage(base, 0);
#pragma unroll 1
    for (int i = 0; i < nit; ++i) {
        if (i + 1 < nit) {
            stage(base + i + 1, (i + 1) & 1);
            // 4 async ops just issued for i+1; loads retire in order, so
            // asynccnt <= 4 means element i's gather has landed in LDS.
            asm volatile("s_wait_asynccnt 0x4" ::: "memory");
        } else {
            asm volatile("s_wait_asynccnt 0x0" ::: "memory");
        }
        compute(base + i, i & 1);
        // LDS reads of this buffer fully retired before it is re-staged (i+2)
        asm volatile("s_wait_dscnt 0x0" ::: "memory");
    }
}

extern "C" void kernel_launch(void* const* d_in, const int* in_sizes, int n_in,
                              void* d_out, int out_size, void* d_ws, size_t ws_size,
                              hipStream_t stream) {
    (void)n_in; (void)out_size; (void)d_ws; (void)ws_size;
    const int*   x       = (const int*)  d_in[0];
    const float* table   = (const float*)d_in[1];
    const float* weights = (const float*)d_in[2];
    const int*   a0      = (const int*)  d_in[3];
    const int*   b0      = (const int*)  d_in[4];
    const int*   a1      = (const int*)  d_in[5];
    const int*   b1      = (const int*)  d_in[6];
    float* out = (float*)d_out;

    const int batch = in_sizes[0];                       // 524,288
    const long total_waves = ((long)batch + NITER - 1) / NITER;
    const int  blocks = (int)((total_waves + WPB - 1) / WPB);  // 8192

    hipLaunchKernelGGL(hashemb_gather_kernel,
                       dim3(blocks), dim3(256), 0, stream,
                       x, table, weights, a0, b0, a1, b1, out, batch);
}